// con_loss_28406913696465
// MI455X (gfx1250) — compile-verified
//
#include <hip/hip_runtime.h>
#include <math.h>

typedef float v2f __attribute__((ext_vector_type(2)));
typedef float v8f __attribute__((ext_vector_type(8)));

#define NIMG 16
#define WDIM 512
#define HPAD 528                 // padded-row buffer height (>=522, multiple of 16)
#define IMGPIX (WDIM * WDIM)     // 262144
#define PIX (NIMG * IMGPIX)      // 4194304
#define RED_BLOCKS 1024
#define EPS 1e-8f

__device__ __forceinline__ int mir512(int q) { q = q < 0 ? -q : q; return q > 511 ? 1022 - q : q; }
__device__ __forceinline__ float band01(int d) { return (d >= 0 && d <= 10) ? 1.0f : 0.0f; }

__device__ __forceinline__ float block_reduce_sum(float v, float* sm) {
    int t = threadIdx.x;
    sm[t] = v; __syncthreads();
    for (int s = 128; s > 0; s >>= 1) { if (t < s) sm[t] += sm[t + s]; __syncthreads(); }
    float r = sm[0]; __syncthreads();
    return r;
}
__device__ __forceinline__ float block_reduce_min(float v, float* sm) {
    int t = threadIdx.x;
    sm[t] = v; __syncthreads();
    for (int s = 128; s > 0; s >>= 1) { if (t < s) sm[t] = fminf(sm[t], sm[t + s]); __syncthreads(); }
    float r = sm[0]; __syncthreads();
    return r;
}
__device__ __forceinline__ float block_reduce_max(float v, float* sm) {
    int t = threadIdx.x;
    sm[t] = v; __syncthreads();
    for (int s = 128; s > 0; s >>= 1) { if (t < s) sm[t] = fmaxf(sm[t], sm[t + s]); __syncthreads(); }
    float r = sm[0]; __syncthreads();
    return r;
}

// ---------------- reductions ----------------
__global__ __launch_bounds__(256) void k_sum1(const float* __restrict__ x, int n, float* __restrict__ part) {
    __shared__ float sm[256];
    float s = 0.f;
    for (int i = blockIdx.x * blockDim.x + threadIdx.x; i < n; i += gridDim.x * blockDim.x) s += x[i];
    float r = block_reduce_sum(s, sm);
    if (threadIdx.x == 0) part[blockIdx.x] = r;
}

__global__ __launch_bounds__(256) void k_sum2(const float* __restrict__ part, float* __restrict__ out, float scale) {
    __shared__ float sm[256];
    int t = threadIdx.x;
    float s = part[t] + part[t + 256] + part[t + 512] + part[t + 768];
    float r = block_reduce_sum(s, sm);
    if (t == 0) out[0] = r * scale;
}

// y = relu(x - mean[0]); also accumulate min/max partials of y
__global__ __launch_bounds__(256) void k_relu_mm1(const float* __restrict__ x, float* __restrict__ y,
                                                  const float* __restrict__ meanp, int n,
                                                  float* __restrict__ pmin, float* __restrict__ pmax) {
    __shared__ float sm[256];
    const float m = meanp[0];
    float lo = 3.4e38f, hi = -3.4e38f;
    for (int i = blockIdx.x * blockDim.x + threadIdx.x; i < n; i += gridDim.x * blockDim.x) {
        float v = x[i] - m; v = v > 0.f ? v : 0.f;
        y[i] = v;
        lo = fminf(lo, v); hi = fmaxf(hi, v);
    }
    float rlo = block_reduce_min(lo, sm);
    float rhi = block_reduce_max(hi, sm);
    if (threadIdx.x == 0) { pmin[blockIdx.x] = rlo; pmax[blockIdx.x] = rhi; }
}

__global__ __launch_bounds__(256) void k_mm1(const float* __restrict__ x, int n,
                                             float* __restrict__ pmin, float* __restrict__ pmax) {
    __shared__ float sm[256];
    float lo = 3.4e38f, hi = -3.4e38f;
    for (int i = blockIdx.x * blockDim.x + threadIdx.x; i < n; i += gridDim.x * blockDim.x) {
        float v = x[i];
        lo = fminf(lo, v); hi = fmaxf(hi, v);
    }
    float rlo = block_reduce_min(lo, sm);
    float rhi = block_reduce_max(hi, sm);
    if (threadIdx.x == 0) { pmin[blockIdx.x] = rlo; pmax[blockIdx.x] = rhi; }
}

__global__ __launch_bounds__(256) void k_mm2(const float* __restrict__ pmin, const float* __restrict__ pmax,
                                             float* __restrict__ lo, float* __restrict__ hi) {
    __shared__ float sm[256];
    int t = threadIdx.x;
    float a = fminf(fminf(pmin[t], pmin[t + 256]), fminf(pmin[t + 512], pmin[t + 768]));
    float b = fmaxf(fmaxf(pmax[t], pmax[t + 256]), fmaxf(pmax[t + 512], pmax[t + 768]));
    float ra = block_reduce_min(a, sm);
    float rb = block_reduce_max(b, sm);
    if (t == 0) { lo[0] = ra; hi[0] = rb; }
}

// ---------------- pointwise ----------------
__device__ __forceinline__ float at0(const float* p, int r, int c) {
    return (r < 0 || r > 511 || c < 0 || c > 511) ? 0.f : p[r * WDIM + c];
}

__global__ __launch_bounds__(256) void k_sobel(const float* __restrict__ x, float* __restrict__ y) {
    int i = blockIdx.x * 256 + threadIdx.x;
    int c = i & 511, r = (i >> 9) & 511, b = i >> 18;
    const float* p = x + (size_t)b * IMGPIX;
    float tl = at0(p, r - 1, c - 1), tc = at0(p, r - 1, c), tr = at0(p, r - 1, c + 1);
    float ml = at0(p, r, c - 1), mr = at0(p, r, c + 1);
    float bl = at0(p, r + 1, c - 1), bc = at0(p, r + 1, c), br = at0(p, r + 1, c + 1);
    float gx = (tr - tl) + 2.f * (mr - ml) + (br - bl);
    float gy = (bl - tl) + 2.f * (bc - tc) + (br - tr);
    y[i] = tanhf(fabsf(gx) + fabsf(gy));
}

__global__ __launch_bounds__(256) void k_sqdiff(const float* __restrict__ a, const float* __restrict__ b,
                                                float* __restrict__ y) {
    int i = blockIdx.x * 256 + threadIdx.x;
    float d = a[i] - b[i];
    y[i] = d * d;
}

// ---------------- WMMA box filter: horizontal pass ----------------
// src: 16x512x512 (reflect indexed in both dims). dst: 16x528x512, rows 0..521 valid.
// mode: 0 identity, 1 apply (x - sc[loIdx]) / (sc[loIdx+1] - sc[loIdx]) before summing.
__global__ __launch_bounds__(256) void k_boxh(const float* __restrict__ src, float* __restrict__ dst,
                                              const float* __restrict__ sc, int loIdx, int mode) {
    const int lane = threadIdx.x & 31;
    const int wv = threadIdx.x >> 5;
    const int tile = blockIdx.x * 8 + wv;
    const int img = tile / (33 * 32);
    const int rem = tile % (33 * 32);
    const int r0 = (rem / 32) * 16;   // padded-row-space base [0,528)
    const int c0 = (rem % 32) * 16;   // output col base [0,512)
    const int n = lane & 15;
    const int half = lane >> 4;

    float lo = 0.f, s = 1.f;
    if (mode) { lo = sc[loIdx]; s = 1.0f / (sc[loIdx + 1] - lo); }

    const int rs = mir512(r0 + n - 5);
    const float* srow = src + (size_t)img * IMGPIX + (size_t)rs * WDIM;

    v8f acc = {0.f, 0.f, 0.f, 0.f, 0.f, 0.f, 0.f, 0.f};
#pragma unroll
    for (int w = 0; w < 8; ++w) {
        const int t0 = 4 * w + 2 * half;
        const int cp = c0 + t0;
        v2f a;
        a.x = srow[mir512(cp - 5)];
        a.y = srow[mir512(cp - 4)];
        if (mode) { a.x = (a.x - lo) * s; a.y = (a.y - lo) * s; }
        v2f b;
        b.x = band01(t0 - n);
        b.y = band01(t0 + 1 - n);
        acc = __builtin_amdgcn_wmma_f32_16x16x4_f32(false, a, false, b, (short)0, acc, false, false);
    }
    float* drow = dst + (size_t)img * (HPAD * WDIM) + (size_t)(r0 + 8 * half) * WDIM + c0 + n;
#pragma unroll
    for (int g = 0; g < 8; ++g) drow[(size_t)g * WDIM] = acc[g];
}

// ---------------- WMMA box filter: vertical pass ----------------
// th: 16x528x512 (rows 0..521 valid). dst: 16x512x512.
// mode: 0 raw sum, 1 divide by 121 (avg pool), 2 sqrt(sum + EPS).
__global__ __launch_bounds__(256) void k_boxv(const float* __restrict__ th, float* __restrict__ dst, int mode) {
    const int lane = threadIdx.x & 31;
    const int wv = threadIdx.x >> 5;
    const int tile = blockIdx.x * 8 + wv;
    const int img = tile >> 10;          // / 1024
    const int rem = tile & 1023;
    const int r0 = (rem >> 5) * 16;
    const int c0 = (rem & 31) * 16;
    const int n = lane & 15;
    const int half = lane >> 4;

    const float* tb = th + (size_t)img * (HPAD * WDIM) + c0 + n;

    v8f acc = {0.f, 0.f, 0.f, 0.f, 0.f, 0.f, 0.f, 0.f};
#pragma unroll
    for (int w = 0; w < 8; ++w) {
        const int t0 = 4 * w + 2 * half;
        v2f a;                           // banded Wv, per-lane constant
        a.x = band01(t0 - n);
        a.y = band01(t0 + 1 - n);
        v2f b;                           // Th tile rows (coalesced across lanes)
        b.x = tb[(size_t)(r0 + t0) * WDIM];
        b.y = tb[(size_t)(r0 + t0 + 1) * WDIM];
        acc = __builtin_amdgcn_wmma_f32_16x16x4_f32(false, a, false, b, (short)0, acc, false, false);
    }
    float* drow = dst + (size_t)img * IMGPIX + (size_t)(r0 + 8 * half) * WDIM + c0 + n;
#pragma unroll
    for (int g = 0; g < 8; ++g) {
        float v = acc[g];
        if (mode == 1) v *= (1.0f / 121.0f);
        else if (mode == 2) v = sqrtf(v + EPS);
        drow[(size_t)g * WDIM] = v;
    }
}

// ---------------- combine weight maps ----------------
__global__ __launch_bounds__(256) void k_combine(const float* __restrict__ pbv, const float* __restrict__ pbi,
                                                 const float* __restrict__ pgv, const float* __restrict__ pgi,
                                                 const float* __restrict__ sc,
                                                 float* __restrict__ outa, float* __restrict__ outb,
                                                 float* __restrict__ psa, float* __restrict__ psb) {
    __shared__ float sm[256];
    const float lo6 = sc[6],  s6  = 1.0f / (sc[7]  - sc[6]);
    const float lo8 = sc[8],  s8  = 1.0f / (sc[9]  - sc[8]);
    const float lo10 = sc[10], s10 = 1.0f / (sc[11] - sc[10]);
    const float lo12 = sc[12], s12 = 1.0f / (sc[13] - sc[12]);
    float sa = 0.f, sb = 0.f;
    for (int i = blockIdx.x * blockDim.x + threadIdx.x; i < PIX; i += gridDim.x * blockDim.x) {
        float bv = expf((pbv[i] - lo6) * s6);
        float bi = expf((pbi[i] - lo8) * s8);
        float gv = expf((pgv[i] - lo10) * s10);
        float gi = expf((pgi[i] - lo12) * s12);
        float iB = 1.0f / (bv + bi);
        float iG = 1.0f / (gv + gi);
        float av = 0.4f * bv * iB + 0.6f * gv * iG;
        float bw = 0.4f * bi * iB + 0.6f * gi * iG;
        outa[i] = av; outb[i] = bw;
        sa += av; sb += bw;
    }
    float ra = block_reduce_sum(sa, sm);
    float rb = block_reduce_sum(sb, sm);
    if (threadIdx.x == 0) { psa[blockIdx.x] = ra; psb[blockIdx.x] = rb; }
}

// ---------------- final per-image weighted mean ----------------
__global__ __launch_bounds__(256) void k_final(const float* __restrict__ ya, const float* __restrict__ yb,
                                               const float* __restrict__ n1, const float* __restrict__ n2,
                                               const float* __restrict__ sc, float* __restrict__ out) {
    __shared__ float sm[256];
    const int b = blockIdx.x;
    const float loa = sc[16], sa = 1.0f / (sc[17] - sc[16]);
    const float lob = sc[18], sb = 1.0f / (sc[19] - sc[18]);
    const size_t off = (size_t)b * IMGPIX;
    float s = 0.f;
    for (int i = threadIdx.x; i < IMGPIX; i += 256) {
        float mv = (ya[off + i] - loa) * sa;
        float mi = (yb[off + i] - lob) * sb;
        s += mv * n1[off + i] + mi * n2[off + i];
    }
    float r = block_reduce_sum(s, sm);
    if (threadIdx.x == 0) out[b] = r * (1.0f / (float)IMGPIX);
}

extern "C" void kernel_launch(void* const* d_in, const int* in_sizes, int n_in,
                              void* d_out, int out_size, void* d_ws, size_t ws_size,
                              hipStream_t stream) {
    const float* vis = (const float*)d_in[0];
    const float* ir  = (const float*)d_in[1];
    const float* fus = (const float*)d_in[2];
    float* out = (float*)d_out;

    const size_t IMG  = (size_t)PIX;
    const size_t THSZ = (size_t)NIMG * HPAD * WDIM;
    float* Wb = (float*)d_ws;
    float* B0 = Wb;
    float* B1 = B0 + IMG;
    float* B2 = B1 + IMG;
    float* B3 = B2 + IMG;
    float* B4 = B3 + IMG;
    float* B5 = B4 + IMG;
    float* TH = B5 + IMG;
    float* p0 = TH + THSZ;
    float* p1 = p0 + 1024;
    float* sc = p1 + 1024;

    const int MAPB = PIX / 256;         // 16384
    const int HB = NIMG * 33 * 32 / 8;  // 2112 blocks (8 waves/block, 1 tile/wave)
    const int VB = NIMG * 32 * 32 / 8;  // 2048

    // global means of vis, ir -> sc[0], sc[1]
    k_sum1<<<RED_BLOCKS, 256, 0, stream>>>(vis, PIX, p0);
    k_sum2<<<1, 256, 0, stream>>>(p0, sc + 0, 1.0f / (float)PIX);
    k_sum1<<<RED_BLOCKS, 256, 0, stream>>>(ir, PIX, p0);
    k_sum2<<<1, 256, 0, stream>>>(p0, sc + 1, 1.0f / (float)PIX);

    // bright(vis): res -> B0, minmax sc[2..3]; normalized box avg -> B1; minmax sc[6..7]
    k_relu_mm1<<<RED_BLOCKS, 256, 0, stream>>>(vis, B0, sc + 0, PIX, p0, p1);
    k_mm2<<<1, 256, 0, stream>>>(p0, p1, sc + 2, sc + 3);
    k_boxh<<<HB, 256, 0, stream>>>(B0, TH, sc, 2, 1);
    k_boxv<<<VB, 256, 0, stream>>>(TH, B1, 1);
    k_mm1<<<RED_BLOCKS, 256, 0, stream>>>(B1, PIX, p0, p1);
    k_mm2<<<1, 256, 0, stream>>>(p0, p1, sc + 6, sc + 7);

    // bright(ir): pooled -> B2; minmax sc[8..9]
    k_relu_mm1<<<RED_BLOCKS, 256, 0, stream>>>(ir, B0, sc + 1, PIX, p0, p1);
    k_mm2<<<1, 256, 0, stream>>>(p0, p1, sc + 4, sc + 5);
    k_boxh<<<HB, 256, 0, stream>>>(B0, TH, sc, 4, 1);
    k_boxv<<<VB, 256, 0, stream>>>(TH, B2, 1);
    k_mm1<<<RED_BLOCKS, 256, 0, stream>>>(B2, PIX, p0, p1);
    k_mm2<<<1, 256, 0, stream>>>(p0, p1, sc + 8, sc + 9);

    // grad(vis): pooled -> B3; minmax sc[10..11]
    k_sobel<<<MAPB, 256, 0, stream>>>(vis, B0);
    k_boxh<<<HB, 256, 0, stream>>>(B0, TH, sc, 0, 0);
    k_boxv<<<VB, 256, 0, stream>>>(TH, B3, 1);
    k_mm1<<<RED_BLOCKS, 256, 0, stream>>>(B3, PIX, p0, p1);
    k_mm2<<<1, 256, 0, stream>>>(p0, p1, sc + 10, sc + 11);

    // grad(ir): pooled -> B4; minmax sc[12..13]
    k_sobel<<<MAPB, 256, 0, stream>>>(ir, B0);
    k_boxh<<<HB, 256, 0, stream>>>(B0, TH, sc, 0, 0);
    k_boxv<<<VB, 256, 0, stream>>>(TH, B4, 1);
    k_mm1<<<RED_BLOCKS, 256, 0, stream>>>(B4, PIX, p0, p1);
    k_mm2<<<1, 256, 0, stream>>>(p0, p1, sc + 12, sc + 13);

    // combine -> a in B0, b in B5; means sc[14..15]
    k_combine<<<RED_BLOCKS, 256, 0, stream>>>(B1, B2, B3, B4, sc, B0, B5, p0, p1);
    k_sum2<<<1, 256, 0, stream>>>(p0, sc + 14, 1.0f / (float)PIX);
    k_sum2<<<1, 256, 0, stream>>>(p1, sc + 15, 1.0f / (float)PIX);
    // relu(a - mean_a) in place, minmax sc[16..17]; same for b -> sc[18..19]
    k_relu_mm1<<<RED_BLOCKS, 256, 0, stream>>>(B0, B0, sc + 14, PIX, p0, p1);
    k_mm2<<<1, 256, 0, stream>>>(p0, p1, sc + 16, sc + 17);
    k_relu_mm1<<<RED_BLOCKS, 256, 0, stream>>>(B5, B5, sc + 15, PIX, p0, p1);
    k_mm2<<<1, 256, 0, stream>>>(p0, p1, sc + 18, sc + 19);

    // window L2: n1 -> B2, n2 -> B3 (pooled buffers now free)
    k_sqdiff<<<MAPB, 256, 0, stream>>>(vis, fus, B1);
    k_boxh<<<HB, 256, 0, stream>>>(B1, TH, sc, 0, 0);
    k_boxv<<<VB, 256, 0, stream>>>(TH, B2, 2);
    k_sqdiff<<<MAPB, 256, 0, stream>>>(ir, fus, B1);
    k_boxh<<<HB, 256, 0, stream>>>(B1, TH, sc, 0, 0);
    k_boxv<<<VB, 256, 0, stream>>>(TH, B3, 2);

    // final per-image weighted means -> d_out[16]
    k_final<<<NIMG, 256, 0, stream>>>(B0, B5, B2, B3, sc, out);
}